// MultiHeadedAttention_17489106830130
// MI455X (gfx1250) — compile-verified
//
#include <hip/hip_runtime.h>

typedef __attribute__((ext_vector_type(2))) float v2f;
typedef __attribute__((ext_vector_type(8))) float v8f;

#define B_  4
#define S_  2048
#define D_  1024
#define H_  16
#define HD_ 64

// ---------------------------------------------------------------------------
// QKV projection: dst[b][h][s][hd] = sum_k X[(b,s)][k] * W[k][(h,hd)]
// M=8192, N=1024, K=1024.  WG = 256 thr = 8 waves; WG tile 64(M) x 128(N);
// each wave computes a 32x32 tile as 2x2 WMMA_F32_16X16X4_F32 accumulators.
// ---------------------------------------------------------------------------
__global__ __launch_bounds__(256)
void qkv_gemm(const float* __restrict__ X, const float* __restrict__ W,
              float* __restrict__ dst)
{
    const int lane = threadIdx.x & 31;
    const int wave = threadIdx.x >> 5;
    const int cl   = lane & 15;      // column/row-in-group lane
    const int hi   = lane >> 4;      // 0: lanes 0-15, 1: lanes 16-31
    const int wm   = wave & 1;       // 2 M sub-blocks of 32
    const int wn   = wave >> 1;      // 4 N sub-blocks of 32
    const int row0 = blockIdx.y * 64 + wm * 32;
    const int col0 = blockIdx.x * 128 + wn * 32;

    v8f acc[2][2];
#pragma unroll
    for (int mi = 0; mi < 2; ++mi)
#pragma unroll
        for (int ni = 0; ni < 2; ++ni)
            acc[mi][ni] = (v8f){0.f, 0.f, 0.f, 0.f, 0.f, 0.f, 0.f, 0.f};

    for (int k = 0; k < D_; k += 4) {
        // A fragments (16x4 f32): lane<16 -> {K=k,k+1}, lane>=16 -> {K=k+2,k+3}
        v2f a[2];
#pragma unroll
        for (int mi = 0; mi < 2; ++mi)
            a[mi] = *(const v2f*)(X + (size_t)(row0 + mi * 16 + cl) * D_ + k + hi * 2);
        // B fragments (4x16 f32): v0 = rows {k | k+2}, v1 = rows {k+1 | k+3}
        v2f b[2];
#pragma unroll
        for (int ni = 0; ni < 2; ++ni) {
            const float* wp = W + (size_t)(k + hi * 2) * D_ + col0 + ni * 16 + cl;
            v2f t; t.x = wp[0]; t.y = wp[D_];
            b[ni] = t;
        }
#pragma unroll
        for (int mi = 0; mi < 2; ++mi)
#pragma unroll
            for (int ni = 0; ni < 2; ++ni)
                acc[mi][ni] = __builtin_amdgcn_wmma_f32_16x16x4_f32(
                    false, a[mi], false, b[ni], (short)0, acc[mi][ni], false, false);
    }

    // C/D layout: VGPR r, lanes0-15 -> row r, lanes16-31 -> row 8+r; col = lane&15
#pragma unroll
    for (int mi = 0; mi < 2; ++mi)
#pragma unroll
        for (int ni = 0; ni < 2; ++ni)
#pragma unroll
            for (int r = 0; r < 8; ++r) {
                int row = row0 + mi * 16 + r + hi * 8;   // (b, s)
                int col = col0 + ni * 16 + cl;           // (h, hd)
                int bb = row >> 11, ss = row & (S_ - 1);
                int hh = col >> 6,  hd = col & (HD_ - 1);
                dst[(((size_t)(bb * H_ + hh) * S_ + ss) * HD_) + hd] = acc[mi][ni][r];
            }
}

// ---------------------------------------------------------------------------
// Causal flash attention over [B*H] x [S x HD], HD=64.
// One wave per 16-query tile; online softmax; P->A relayout via LDS.
// V tiles are double-buffered in LDS via async DMA (GLOBAL_LOAD_ASYNC_TO_LDS,
// tracked with ASYNCcnt) so the next tile's V fetch overlaps QK^T + softmax.
// ---------------------------------------------------------------------------
__device__ __forceinline__ void async_copy_vtile(const float* __restrict__ src,
                                                 uint32_t lds_byte_base, int lane)
{
    // 16x64 f32 tile = 4KB; 8 x b128 per-lane async ops, 512B each.
#pragma unroll
    for (int i = 0; i < 8; ++i) {
        uint32_t lds = lds_byte_base + i * 512 + lane * 16;
        const float* g = src + i * 128 + lane * 4;
        asm volatile("global_load_async_to_lds_b128 %0, %1, off"
                     :: "v"(lds), "v"(g) : "memory");
    }
}

__global__ __launch_bounds__(128)
void attn(const float* __restrict__ Q, const float* __restrict__ K,
          const float* __restrict__ V, float* __restrict__ out)
{
    __shared__ float vbuf[4][2][16 * 64];   // per-wave double-buffered V tile
    __shared__ float pbuf[4][16][17];       // per-wave 16x16 P tile (padded)

    const int lane = threadIdx.x & 31;
    const int wave = threadIdx.x >> 5;
    const int cl   = lane & 15;
    const int hi   = lane >> 4;

    const int gw = blockIdx.x * 4 + wave;    // 8192 waves total
    const int qt = gw & 127;                 // 128 query tiles per (b,h)
    const int bh = gw >> 7;                  // 0..63
    const int qm = qt * 16;

    const float* Qp = Q + (size_t)bh * S_ * HD_;
    const float* Kp = K + (size_t)bh * S_ * HD_;
    const float* Vp = V + (size_t)bh * S_ * HD_;

    const uint32_t vlds0 = (uint32_t)(uintptr_t)&vbuf[wave][0][0];
    const uint32_t vlds1 = (uint32_t)(uintptr_t)&vbuf[wave][1][0];

    // Q tile 16x64 as 16 A-fragments (loaded once, b64 per lane per chunk)
    v2f qa[16];
#pragma unroll
    for (int c = 0; c < 16; ++c)
        qa[c] = *(const v2f*)(Qp + (size_t)(qm + cl) * HD_ + c * 4 + hi * 2);

    v8f o[4];
#pragma unroll
    for (int nt = 0; nt < 4; ++nt)
        o[nt] = (v8f){0.f, 0.f, 0.f, 0.f, 0.f, 0.f, 0.f, 0.f};

    float m[8], l[8];
#pragma unroll
    for (int r = 0; r < 8; ++r) { m[r] = -3.0e38f; l[r] = 0.f; }

    // kick off async DMA of the first V tile into buffer 0
    async_copy_vtile(Vp, vlds0, lane);

    for (int kn = 0; kn <= qm; kn += 16) {
        const int  cur      = (kn >> 4) & 1;
        const bool havenext = (kn + 16 <= qm);
        // prefetch next V tile into the other buffer (overlaps QK^T + softmax)
        if (havenext)
            async_copy_vtile(Vp + (size_t)(kn + 16) * HD_,
                             cur ? vlds0 : vlds1, lane);

        // S = Q * K^T  (16 WMMAs over HD)
        v8f s = (v8f){0.f, 0.f, 0.f, 0.f, 0.f, 0.f, 0.f, 0.f};
#pragma unroll
        for (int c = 0; c < 16; ++c) {
            v2f kb = *(const v2f*)(Kp + (size_t)(kn + cl) * HD_ + c * 4 + hi * 2);
            s = __builtin_amdgcn_wmma_f32_16x16x4_f32(
                false, qa[c], false, kb, (short)0, s, false, false);
        }

        const bool diag = (kn == qm);
        float alpha[8];
#pragma unroll
        for (int r = 0; r < 8; ++r) {
            float sv = s[r] * 0.125f;                       // 1/sqrt(64)
            if (diag && (cl > r + hi * 8)) sv = -3.0e38f;   // causal mask
            // row max across the 16 lanes holding this row
            float mx = sv;
            mx = fmaxf(mx, __shfl_xor(mx, 1, 16));
            mx = fmaxf(mx, __shfl_xor(mx, 2, 16));
            mx = fmaxf(mx, __shfl_xor(mx, 4, 16));
            mx = fmaxf(mx, __shfl_xor(mx, 8, 16));
            float mnew = fmaxf(m[r], mx);
            float a    = __expf(m[r] - mnew);
            float pv   = __expf(sv - mnew);
            float rs = pv;
            rs += __shfl_xor(rs, 1, 16);
            rs += __shfl_xor(rs, 2, 16);
            rs += __shfl_xor(rs, 4, 16);
            rs += __shfl_xor(rs, 8, 16);
            l[r] = l[r] * a + rs;
            m[r] = mnew;
            alpha[r] = a;
            pbuf[wave][r + hi * 8][cl] = pv;    // stage P for A-layout reload
        }

        // rescale running output
#pragma unroll
        for (int nt = 0; nt < 4; ++nt)
#pragma unroll
            for (int r = 0; r < 8; ++r)
                o[nt][r] *= alpha[r];

        // wait for the *current* V tile: async loads complete in order, so with
        // 8 prefetch ops still outstanding "<=8" proves the current 8 landed.
        if (havenext)
            asm volatile("s_wait_asynccnt 0x8" ::: "memory");
        else
            asm volatile("s_wait_asynccnt 0x0" ::: "memory");

        // O += P(16x16) * V(16x64): 4 K-chunks x 4 N-tiles, V from LDS
        const float* vt = &vbuf[wave][cur][0];
#pragma unroll
        for (int kc = 0; kc < 4; ++kc) {
            v2f pa;
            pa.x = pbuf[wave][cl][kc * 4 + hi * 2];
            pa.y = pbuf[wave][cl][kc * 4 + hi * 2 + 1];
#pragma unroll
            for (int nt = 0; nt < 4; ++nt) {
                const float* vp = vt + (kc * 4 + hi * 2) * HD_ + nt * 16 + cl;
                v2f vb; vb.x = vp[0]; vb.y = vp[HD_];
                o[nt] = __builtin_amdgcn_wmma_f32_16x16x4_f32(
                    false, pa, false, vb, (short)0, o[nt], false, false);
            }
        }
    }

    // epilogue: normalize by l and scatter to out[b][s][h*64+hd]
    const int bb = bh >> 4, hh = bh & 15;
    float inv[8];
#pragma unroll
    for (int r = 0; r < 8; ++r) inv[r] = 1.0f / l[r];
#pragma unroll
    for (int nt = 0; nt < 4; ++nt)
#pragma unroll
        for (int r = 0; r < 8; ++r) {
            int srow = qm + r + hi * 8;
            int col  = hh * HD_ + nt * 16 + cl;
            out[(size_t)(bb * S_ + srow) * D_ + col] = o[nt][r] * inv[r];
        }
}

// ---------------------------------------------------------------------------
extern "C" void kernel_launch(void* const* d_in, const int* in_sizes, int n_in,
                              void* d_out, int out_size, void* d_ws, size_t ws_size,
                              hipStream_t stream) {
    (void)in_sizes; (void)n_in; (void)out_size; (void)ws_size;
    const float* x  = (const float*)d_in[0];
    const float* Wq = (const float*)d_in[1];
    const float* Wk = (const float*)d_in[2];
    const float* Wv = (const float*)d_in[3];
    float* out = (float*)d_out;

    float* Qws = (float*)d_ws;                         // [B][H][S][HD] each
    float* Kws = Qws + (size_t)B_ * S_ * D_;
    float* Vws = Kws + (size_t)B_ * S_ * D_;

    dim3 ggrid(D_ / 128, (B_ * S_) / 64, 1);           // (8, 128)
    qkv_gemm<<<ggrid, 256, 0, stream>>>(x, Wq, Qws);
    qkv_gemm<<<ggrid, 256, 0, stream>>>(x, Wk, Kws);
    qkv_gemm<<<ggrid, 256, 0, stream>>>(x, Wv, Vws);

    attn<<<(B_ * H_ * (S_ / 16)) / 4, 128, 0, stream>>>(Qws, Kws, Vws, out);
}